// MyTriplet_loss_44650480009710
// MI455X (gfx1250) — compile-verified
//
#include <hip/hip_runtime.h>

#define N 512
#define D 256
#define MARGIN 1.0f

typedef __attribute__((ext_vector_type(2))) float v2f;
typedef __attribute__((ext_vector_type(8))) float v8f;

// ---------------- Kernel 0: row squared norms (sq[i] = G[i,i]) ----------------
__global__ void Triplet_sqnorm(const float* __restrict__ X, float* __restrict__ sq) {
    int i = blockIdx.x * blockDim.x + threadIdx.x;
    if (i < N) {
        const float* r = X + i * D;
        float s = 0.0f;
#pragma unroll 4
        for (int k = 0; k < D; k += 4) {
            float4 v = *(const float4*)(r + k);
            s += v.x * v.x + v.y * v.y + v.z * v.z + v.w * v.w;
        }
        sq[i] = s;
    }
}

// ---------------- Kernel 1: pairwise distances via fp32 WMMA ----------------
// One wave per 16x16 tile of the 512x512 Gram matrix; K-loop of 16x16x4 f32 WMMAs.
__global__ void Triplet_dist_wmma(const float* __restrict__ X,
                                  const float* __restrict__ sq,
                                  float* __restrict__ dist) {
    int wave = blockIdx.x * (blockDim.x >> 5) + (threadIdx.x >> 5);
    int lane = threadIdx.x & 31;
    int ti = wave >> 5;   // 32 tiles per dim
    int tj = wave & 31;

    int m     = lane & 15;            // matrix row index within tile (A) / col (B)
    int kbase = (lane >> 4) << 1;     // lanes 0-15 hold K={0,1}; lanes 16-31 hold K={2,3}

    const float* arow = X + (ti * 16 + m) * D;   // A = X tile rows
    const float* brow = X + (tj * 16 + m) * D;   // B = X^T -> B[k][n] = X[n][k]

    v8f acc = {};
    for (int k0 = 0; k0 < D; k0 += 4) {
        v2f a = *(const v2f*)(arow + k0 + kbase);
        v2f b = *(const v2f*)(brow + k0 + kbase);
        // (neg_a, A, neg_b, B, c_mod, C, reuse_a, reuse_b)
        acc = __builtin_amdgcn_wmma_f32_16x16x4_f32(false, a, false, b,
                                                    (short)0, acc, false, false);
    }

    // C/D layout: VGPR v -> M = v + 8*(lane/16), N = lane%16
    int jcol  = tj * 16 + (lane & 15);
    int ibase = ti * 16 + ((lane >> 4) << 3);
    float sqj = sq[jcol];
#pragma unroll
    for (int v = 0; v < 8; ++v) {
        int irow = ibase + v;
        float val = sq[irow] + sqj - 2.0f * acc[v];
        dist[irow * N + jcol] = sqrtf(fmaxf(val, 1e-16f));
    }
}

// ---------------- Kernel 2: masked triplet accumulation, one block per anchor ----------------
__global__ void Triplet_body(const float* __restrict__ dist,
                             const int* __restrict__ labels,
                             float* __restrict__ psum,
                             unsigned int* __restrict__ pcnt) {
    __shared__ float row[N];
    __shared__ int   lab[N];
    __shared__ float    ssum[256];
    __shared__ unsigned scnt[256];

    int i   = blockIdx.x;
    int tid = threadIdx.x;

    for (int j = tid; j < N; j += 256) {
        row[j] = dist[i * N + j];
        lab[j] = labels[j];
    }
    __syncthreads();

    int li = lab[i];
    float s = 0.0f;
    unsigned c = 0;

    // mask: labels[i]==labels[j] & j!=i & labels[i]!=labels[k]
    // (labels[k]!=li implies k!=i and k!=j automatically)
    for (int j = tid; j < N; j += 256) {
        if (j != i && lab[j] == li) {
            float dij = row[j];
            for (int k = 0; k < N; ++k) {
                if (lab[k] != li) {
                    float t = dij - row[k] + MARGIN;
                    if (t > 0.0f) {
                        s += t;                     // sum of max(t, 0)
                        c += (t > 1e-16f) ? 1u : 0u; // count of > 1e-16
                    }
                }
            }
        }
    }

    ssum[tid] = s;
    scnt[tid] = c;
    __syncthreads();
    for (int off = 128; off > 0; off >>= 1) {
        if (tid < off) {
            ssum[tid] += ssum[tid + off];
            scnt[tid] += scnt[tid + off];
        }
        __syncthreads();
    }
    if (tid == 0) {
        psum[i] = ssum[0];
        pcnt[i] = scnt[0];
    }
}

// ---------------- Kernel 3: deterministic final reduction ----------------
__global__ void Triplet_finalize(const float* __restrict__ psum,
                                 const unsigned int* __restrict__ pcnt,
                                 float* __restrict__ out) {
    __shared__ float    ssum[N];
    __shared__ unsigned scnt[N];
    int tid = threadIdx.x;   // 512 threads
    ssum[tid] = psum[tid];
    scnt[tid] = pcnt[tid];
    __syncthreads();
    for (int off = 256; off > 0; off >>= 1) {
        if (tid < off) {
            ssum[tid] += ssum[tid + off];
            scnt[tid] += scnt[tid + off];
        }
        __syncthreads();
    }
    if (tid == 0) {
        out[0] = ssum[0] / ((float)scnt[0] + 1e-16f);
    }
}

extern "C" void kernel_launch(void* const* d_in, const int* in_sizes, int n_in,
                              void* d_out, int out_size, void* d_ws, size_t ws_size,
                              hipStream_t stream) {
    const float* X      = (const float*)d_in[0];   // [512, 256] fp32
    const int*   labels = (const int*)d_in[1];     // [512] int32
    float*       out    = (float*)d_out;           // scalar

    float*        sq   = (float*)d_ws;             // 512 floats
    float*        dist = sq + N;                   // 512*512 floats
    float*        psum = dist + N * N;             // 512 floats
    unsigned int* pcnt = (unsigned int*)(psum + N);// 512 uints

    Triplet_sqnorm   <<<(N + 255) / 256, 256, 0, stream>>>(X, sq);
    Triplet_dist_wmma<<<128, 256, 0, stream>>>(X, sq, dist);   // 1024 waves = 1024 tiles
    Triplet_body     <<<N, 256, 0, stream>>>(dist, labels, psum, pcnt);
    Triplet_finalize <<<1, N, 0, stream>>>(psum, pcnt, out);
}